// ProceduralLayerOnTheFly_78778290143897
// MI455X (gfx1250) — compile-verified
//
#include <hip/hip_runtime.h>
#include <hip/hip_bf16.h>
#include <stdint.h>

// Problem geometry (fixed by the reference).
#define IN_FEATURES   8192
#define OUT_FEATURES  8192
#define FAN_OUT       4096

// Tiling: 512 blocks x 256 threads (8 wave32s). Each block owns 16 rows and a
// full private 32KB output accumulator in LDS. Weights/targets stream through
// a 4-deep LDS staging ring via async global->LDS copies (ASYNCcnt).
// LDS/block = 32KB acc + 4*8KB staging = 64KB -> 5 blocks per 320KB WGP.
#define THREADS         256
#define ROWS_PER_BLOCK  16
#define NUM_BLOCKS      (IN_FEATURES / ROWS_PER_BLOCK)        // 512
#define CHUNK           1024                                   // elements per staged chunk
#define CHUNKS_PER_ROW  (FAN_OUT / CHUNK)                      // 4
#define TOTAL_CHUNKS    (ROWS_PER_BLOCK * CHUNKS_PER_ROW)      // 64
#define NBUF            4                                      // staging ring depth

typedef float v4f __attribute__((ext_vector_type(4)));
typedef int   v4i __attribute__((ext_vector_type(4)));

// Address-space-qualified v4i, matching the builtin signature revealed by the
// round-2 diagnostic: (AS1 v4i* src, AS3 v4i* dst, imm offset, imm cpol).
typedef __attribute__((address_space(1))) v4i GLv4i;   // global
typedef __attribute__((address_space(3))) v4i LDv4i;   // LDS

// cpol: non-temporal load hint -- the 256MB w/t stream is read exactly once;
// keep L2 for the hot 32KB output region instead.
#define ASYNC_CPOL 1

// ---------------------------------------------------------------------------
// gfx1250 async global->LDS b128 copy (ASYNCcnt-tracked).
// Low 32 bits of a flat shared pointer are the LDS byte address (ISA §10.2:
// the address is truncated to its low 32 bits), so inttoptr into AS(3) is exact.
// ---------------------------------------------------------------------------
__device__ __forceinline__ void async_copy_b128(const void* gsrc, void* ldst) {
#if __has_builtin(__builtin_amdgcn_global_load_async_to_lds_b128)
    __builtin_amdgcn_global_load_async_to_lds_b128(
        (GLv4i*)(uintptr_t)gsrc,
        (LDv4i*)(uint32_t)(uintptr_t)ldst,
        /*offset=*/0, /*cpol=*/ASYNC_CPOL);
#else
    uint32_t loff = (uint32_t)(uintptr_t)ldst;   // LDS byte address
    uint64_t ga   = (uint64_t)(uintptr_t)gsrc;   // 64-bit global address
    asm volatile("global_load_async_to_lds_b128 %0, %1, off"
                 :: "v"(loff), "v"(ga) : "memory");
#endif
}

#if __has_builtin(__builtin_amdgcn_s_wait_asynccnt)
#define WAIT_ASYNC(n) __builtin_amdgcn_s_wait_asynccnt(n)
#else
#define WAIT_ASYNC(n) asm volatile("s_wait_asynccnt %0" :: "i"(n) : "memory")
#endif

// ---------------------------------------------------------------------------
// Prologue: zero d_out (harness poisons it with 0xAA before timing).
// ---------------------------------------------------------------------------
__global__ __launch_bounds__(THREADS) void zero_out_kernel(float* __restrict__ out) {
    const int i = blockIdx.x * THREADS + threadIdx.x;   // float4 index
    v4f z = (v4f)(0.0f);
    *reinterpret_cast<v4f*>(out + (size_t)i * 4) = z;
}

// ---------------------------------------------------------------------------
// Main: async-staged, LDS-privatized scatter-add.
// ---------------------------------------------------------------------------
__global__ __launch_bounds__(THREADS) void scatter_add_kernel(
    const float* __restrict__ x,       // [IN_FEATURES]
    const float* __restrict__ w,       // [IN_FEATURES, FAN_OUT]
    const int*   __restrict__ tg,      // [IN_FEATURES, FAN_OUT]
    float*       __restrict__ out)     // [OUT_FEATURES]
{
    __shared__ __align__(16) float acc[OUT_FEATURES];       // 32 KB accumulator
    __shared__ __align__(16) float wbuf[NBUF][CHUNK];       // 4 x 4 KB
    __shared__ __align__(16) int   tbuf[NBUF][CHUNK];       // 4 x 4 KB

    const int tid  = threadIdx.x;
    const int row0 = blockIdx.x * ROWS_PER_BLOCK;
    const float* __restrict__ wblk = w  + (size_t)row0 * FAN_OUT;
    const int*   __restrict__ tblk = tg + (size_t)row0 * FAN_OUT;

    // Thread-private staging partition: thread owns float4 slot [tid*4 .. tid*4+3]
    // of each ring buffer and consumes exactly what it staged -> no barriers in
    // the steady-state loop; ordering is pure in-order ASYNCcnt retirement.
    auto stage = [&](int c, int b) {
        const size_t off = (size_t)c * CHUNK + (size_t)tid * 4;
        async_copy_b128(wblk + off, &wbuf[b][tid * 4]);
        async_copy_b128(tblk + off, &tbuf[b][tid * 4]);
    };
    auto consume = [&](int c, float xv) {
        const int b = c & (NBUF - 1);
        v4f wv = *reinterpret_cast<const v4f*>(&wbuf[b][tid * 4]);   // ds_load_b128
        v4i tv = *reinterpret_cast<const v4i*>(&tbuf[b][tid * 4]);
        atomicAdd(&acc[tv.x], xv * wv.x);        // ds_add_f32 (no-return)
        atomicAdd(&acc[tv.y], xv * wv.y);
        atomicAdd(&acc[tv.z], xv * wv.z);
        atomicAdd(&acc[tv.w], xv * wv.w);
    };

    // Prime the ring: 3 chunks (6 async ops) in flight before first consume.
    stage(0, 0);
    stage(1, 1);
    stage(2, 2);

    // Zero the accumulator while the first chunks are in flight (ds_store_b128 x8).
#pragma unroll
    for (int k = 0; k < OUT_FEATURES / (THREADS * 4); ++k) {
        *reinterpret_cast<v4f*>(&acc[(k * THREADS + tid) * 4]) = (v4f)(0.0f);
    }
    __syncthreads();                    // acc zeroed before any ds_add_f32

    // Steady state: rows 0..14 (x loaded once per row), 4 chunks per row.
#pragma unroll 1
    for (int r = 0; r < ROWS_PER_BLOCK - 1; ++r) {
        const float xv = x[row0 + r];   // block-uniform, cache-hot
#pragma unroll
        for (int q = 0; q < CHUNKS_PER_ROW; ++q) {
            const int c = r * CHUNKS_PER_ROW + q;
            stage(c + 3, (c + 3) & (NBUF - 1));   // keep 3 chunks ahead
            WAIT_ASYNC(6);                        // in-order: chunk c retired
            consume(c, xv);
        }
    }

    // Last row: stage final chunk, then drain the pipeline.
    {
        const float xv = x[row0 + ROWS_PER_BLOCK - 1];
        stage(TOTAL_CHUNKS - 1, (TOTAL_CHUNKS - 1) & (NBUF - 1));
        WAIT_ASYNC(6); consume(TOTAL_CHUNKS - 4, xv);
        WAIT_ASYNC(4); consume(TOTAL_CHUNKS - 3, xv);
        WAIT_ASYNC(2); consume(TOTAL_CHUNKS - 2, xv);
        WAIT_ASYNC(0); consume(TOTAL_CHUNKS - 1, xv);
    }

    __syncthreads();

    // Flush private accumulator: 8192 device-scope float atomics per block
    // (4.2M total, negligible vs. the 268MB streamed in).
#pragma unroll
    for (int k = 0; k < OUT_FEATURES / THREADS; ++k) {
        const int o = tid + k * THREADS;
        unsafeAtomicAdd(&out[o], acc[o]);        // global_atomic_add_f32
    }
}

// ---------------------------------------------------------------------------
// Harness entry point.
//   d_in[0] = x        (8192 f32)
//   d_in[1] = weights  (8192*4096 f32)
//   d_in[2] = targets  (8192*4096 i32)
//   d_out   = out      (8192 f32)
// ---------------------------------------------------------------------------
extern "C" void kernel_launch(void* const* d_in, const int* in_sizes, int n_in,
                              void* d_out, int out_size, void* d_ws, size_t ws_size,
                              hipStream_t stream) {
    (void)in_sizes; (void)n_in; (void)d_ws; (void)ws_size; (void)out_size;

    const float* x   = reinterpret_cast<const float*>(d_in[0]);
    const float* w   = reinterpret_cast<const float*>(d_in[1]);
    const int*   tg  = reinterpret_cast<const int*>(d_in[2]);
    float*       out = reinterpret_cast<float*>(d_out);

    zero_out_kernel<<<(OUT_FEATURES / 4) / THREADS, THREADS, 0, stream>>>(out);
    scatter_add_kernel<<<NUM_BLOCKS, THREADS, 0, stream>>>(x, w, tg, out);
}